// CombinedLoss_64441689309705
// MI455X (gfx1250) — compile-verified
//
#include <hip/hip_runtime.h>
#include <hip/hip_bf16.h>
#include <cstdint>

#ifndef __has_builtin
#define __has_builtin(x) 0
#endif

#define M_ANN   512
#define N_TOTAL 114688
#define INF_F   100000000.0f

#if __has_builtin(__builtin_amdgcn_global_load_async_to_lds_b128)
#define USE_ASYNC_LDS 1
#else
#define USE_ASYNC_LDS 0
#endif

typedef int v4i __attribute__((ext_vector_type(4)));
typedef __attribute__((address_space(1))) v4i* g_v4i_ptr;
typedef __attribute__((address_space(3))) v4i* l_v4i_ptr;

__device__ __forceinline__ void async_wait_all() {
#if __has_builtin(__builtin_amdgcn_s_wait_asynccnt)
    __builtin_amdgcn_s_wait_asynccnt(0);
#else
    asm volatile("s_wait_asynccnt 0" ::: "memory");
#endif
}

template <int LEVEL>
__global__ __launch_bounds__(256) void assign_level_kernel(
    const float* __restrict__ ann,      // [512,3] (l, r, class) sorted by l
    const float* __restrict__ anchors,  // [n] anchor positions
    float* __restrict__ out)            // [12 * N_TOTAL] concatenated outputs
{
    constexpr float  STRIDE     = (float)(1 << LEVEL);
    constexpr float  INV_STRIDE = 1.0f / STRIDE;
    constexpr double TR  = 22050.0 / 256.0;
    constexpr double LOd = (LEVEL == 0) ? -1.0 : (LEVEL == 1 ? 0.45 : 0.9);
    constexpr double HId = (LEVEL == 0) ? 0.45 : (LEVEL == 1 ? 0.9  : 1000.0);
    constexpr float  LOTR = (float)(LOd * TR);   // matches JAX f64->f32 scalar promotion
    constexpr float  HITR = (float)(HId * TR);
    constexpr int    OFF  = (LEVEL == 0) ? 0 : (LEVEL == 1 ? 65536 : 98304);
    constexpr int    N    = N_TOTAL;

    __shared__ float4 raw4[M_ANN * 3 / 4];          // raw annotation table (6 KB)
    __shared__ float  l_s[M_ANN], r_s[M_ANN], rmin_s[M_ANN], area_s[M_ANN];
    float* raw = (float*)raw4;

    const int tid = threadIdx.x;

    // ---- Stage the annotation table into LDS (async DMA path on gfx1250) ----
#if USE_ASYNC_LDS
    {
        const float4* g4 = (const float4*)ann;
        for (int t = tid; t < M_ANN * 3 / 4; t += 256) {
            void* gpv = (void*)const_cast<float4*>(g4 + t);
            g_v4i_ptr gp = (g_v4i_ptr)gpv;
            l_v4i_ptr lp = (l_v4i_ptr)(void*)(raw4 + t);
            __builtin_amdgcn_global_load_async_to_lds_b128(gp, lp, 0, 0);
        }
        async_wait_all();
    }
#else
    {
        const float4* g4 = (const float4*)ann;
        for (int t = tid; t < M_ANN * 3 / 4; t += 256) raw4[t] = g4[t];
    }
#endif
    __syncthreads();

    // ---- Derived per-annotation arrays (computed once per block) ----
    for (int j = tid; j < M_ANN; j += 256) {
        float l = raw[3 * j + 0];
        float r = raw[3 * j + 1];
        float c = raw[3 * j + 2];
        float radius = (c == 0.0f) ? 4.5f : ((c == 1.0f) ? 2.5f : 0.0f);
        l_s[j]    = l;
        r_s[j]    = r;
        rmin_s[j] = fminf(r, l + radius * STRIDE);   // min(r, radius_limit)
        area_s[j] = r - l;
    }
    __syncthreads();

    const int   i = blockIdx.x * 256 + tid;   // local anchor index (grid is exact)
    const float a = anchors[i];

    // ---- Window over sorted starts: valid j needs l[j] <= a and
    //      l[j] >= a - radius*stride >= a - 4.5*stride. Binary search both ends.
    int lo = 0, hi = M_ANN;
    while (lo < hi) { int mid = (lo + hi) >> 1; if (l_s[mid] <= a) lo = mid + 1; else hi = mid; }
    const int jhi = lo;                               // first index with l > a
    const float wlo = a - 4.5f * STRIDE - 1.0f;       // 1.0 margin >> ulp@65536
    lo = 0; hi = jhi;
    while (lo < hi) { int mid = (lo + hi) >> 1; if (l_s[mid] < wlo) lo = mid + 1; else hi = mid; }

    // ---- Ascending scan preserves argmin first-occurrence tie-break;
    //      empty/invalid window leaves bidx = 0 (matching argmin over all-INF).
    float best = INF_F;
    int   bidx = 0;
    for (int j = lo; j < jhi; ++j) {
        float lj = l_s[j];
        bool ok = (a >= lj) && (a <= rmin_s[j]);
        float m = fmaxf(a - lj, r_s[j] - a);
        ok = ok && (m >= LOTR) && (m <= HITR);
        float cand = ok ? area_s[j] : INF_F;
        if (cand < best) { best = cand; bidx = j; }
    }

    const bool  pos  = (best != INF_F);
    const float lb   = l_s[bidx];
    const float rb   = r_s[bidx];
    const float cb   = raw[3 * bidx + 2];
    const float cOut = pos ? cb : 0.0f;
    const float lst  = a - lb;
    const float rst  = rb - a;

    const int row = OFF + i;
    out[row]                  = pos ? 1.0f : 0.0f;   // pos
    out[N      + 3 * row + 0] = lb;                  // assigned
    out[N      + 3 * row + 1] = rb;
    out[N      + 3 * row + 2] = cOut;
    out[4 * N  + 3 * row + 0] = lb * INV_STRIDE;     // normalized (exact: /2^k)
    out[4 * N  + 3 * row + 1] = rb * INV_STRIDE;
    out[4 * N  + 3 * row + 2] = cOut;
    out[7 * N  + row]         = lst;                 // l_star
    out[8 * N  + row]         = rst;                 // r_star
    out[9 * N  + row]         = lst * INV_STRIDE;    // nl_star
    out[10 * N + row]         = rst * INV_STRIDE;    // nr_star
    out[11 * N + row]         = (float)(LEVEL + 1);  // levels
}

extern "C" void kernel_launch(void* const* d_in, const int* in_sizes, int n_in,
                              void* d_out, int out_size, void* d_ws, size_t ws_size,
                              hipStream_t stream) {
    (void)in_sizes; (void)n_in; (void)out_size; (void)d_ws; (void)ws_size;
    const float* ann = (const float*)d_in[0];
    const float* a0  = (const float*)d_in[1];
    const float* a1  = (const float*)d_in[2];
    const float* a2  = (const float*)d_in[3];
    float* out = (float*)d_out;

    assign_level_kernel<0><<<65536 / 256, 256, 0, stream>>>(ann, a0, out);
    assign_level_kernel<1><<<32768 / 256, 256, 0, stream>>>(ann, a1, out);
    assign_level_kernel<2><<<16384 / 256, 256, 0, stream>>>(ann, a2, out);
}